// MambaEncoder_1159641170573
// MI455X (gfx1250) — compile-verified
//
#include <hip/hip_runtime.h>
#include <hip/hip_bf16.h>
#include <math.h>

#define D_MODEL 768
#define D_INNER 1536
#define D_STATE 16
#define DT_RANK 48
#define D_CONV  4
#define DEPTH   4
#define BB      2
#define LL      1024
#define M_ROWS  (BB * LL)   /* 2048 */

typedef float v2f __attribute__((ext_vector_type(2)));
typedef float v8f __attribute__((ext_vector_type(8)));

// ---------------------------------------------------------------------------
// GEMM: C[M,N] = A[M,K] * W[N,K]^T   (all fp32, row-major)
// One wave computes a (16*MT)(M) x (16*NSUB)(N) tile with
// V_WMMA_F32_16X16X4_F32, stepping K by 4. Grids are launched exactly, so
// there is NO control flow in the K loop: per step we do MT a-loads +
// NSUB b-loads (grouped, one wait) followed by MT*NSUB back-to-back WMMAs.
//
// Fragment mapping (wave32, ISA 7.12.2):
//  A 16x4:  lanes 0-15 -> row M=lane,    K = {k, k+1}
//           lanes 16-31-> row M=lane-16, K = {k+2, k+3}
//  B 4x16:  mirrored: N = lane%16, same K split by lane half
//  C/D:     VGPR r: lanes 0-15 -> M=r, lanes 16-31 -> M=r+8, N=lane%16
// ---------------------------------------------------------------------------
template <int NSUB, int MT>
__global__ __launch_bounds__(32)
void gemm_wmma_f32(const float* __restrict__ A, const float* __restrict__ W,
                   float* __restrict__ C, int K, int lda, int ldw, int ldc)
{
    const int lane  = threadIdx.x & 31;
    const int half  = lane >> 4;
    const int l16   = lane & 15;
    const int mBase = blockIdx.y * (16 * MT);
    const int nBase = blockIdx.x * (16 * NSUB);
    const int koff  = half * 2;

    const float* aRow[MT];
#pragma unroll
    for (int mt = 0; mt < MT; ++mt)
        aRow[mt] = A + (size_t)(mBase + mt * 16 + l16) * lda + koff;

    const float* wRow[NSUB];
#pragma unroll
    for (int s = 0; s < NSUB; ++s)
        wRow[s] = W + (size_t)(nBase + s * 16 + l16) * ldw + koff;

    v8f acc[MT][NSUB];
#pragma unroll
    for (int mt = 0; mt < MT; ++mt)
#pragma unroll
        for (int s = 0; s < NSUB; ++s)
            acc[mt][s] = (v8f){0.f,0.f,0.f,0.f,0.f,0.f,0.f,0.f};

#pragma unroll 2
    for (int k = 0; k < K; k += 4) {
        v2f a[MT], b[NSUB];
#pragma unroll
        for (int mt = 0; mt < MT; ++mt)
            a[mt] = *(const v2f*)(aRow[mt] + k);
#pragma unroll
        for (int s = 0; s < NSUB; ++s)
            b[s] = *(const v2f*)(wRow[s] + k);
#pragma unroll
        for (int s = 0; s < NSUB; ++s)
#pragma unroll
            for (int mt = 0; mt < MT; ++mt)
                acc[mt][s] = __builtin_amdgcn_wmma_f32_16x16x4_f32(
                                 false, a[mt], false, b[s], (short)0,
                                 acc[mt][s], false, false);
    }

#pragma unroll
    for (int mt = 0; mt < MT; ++mt)
#pragma unroll
        for (int s = 0; s < NSUB; ++s) {
            const int col = nBase + s * 16 + l16;
#pragma unroll
            for (int r = 0; r < 8; ++r) {
                const int row = mBase + mt * 16 + r + 8 * half;
                C[(size_t)row * ldc + col] = acc[mt][s][r];
            }
        }
}

// ---------------------------------------------------------------------------
// Causal depthwise conv (width 4, left pad 3) + bias + SiLU.
// Input = first half of xz (stride 2*D_INNER); output u (stride D_INNER).
// ---------------------------------------------------------------------------
__global__ __launch_bounds__(256)
void conv_silu(const float* __restrict__ xz, const float* __restrict__ cw,
               const float* __restrict__ cb, float* __restrict__ u)
{
    const int tid = blockIdx.x * blockDim.x + threadIdx.x;
    if (tid >= BB * LL * D_INNER) return;
    const int d = tid % D_INNER;
    const int l = (tid / D_INNER) % LL;
    const int b = tid / (D_INNER * LL);

    float acc = cb[d];
#pragma unroll
    for (int j = 0; j < D_CONV; ++j) {
        const int t = l - (D_CONV - 1) + j;
        if (t >= 0)
            acc += cw[d * D_CONV + j] *
                   xz[((size_t)(b * LL + t)) * (2 * D_INNER) + d];
    }
    u[(size_t)(b * LL + l) * D_INNER + d] = acc / (1.f + __expf(-acc));
}

// ---------------------------------------------------------------------------
// Fused selective scan: softplus(dt + b), h = exp(dt*A)*h + dt*u*B,
// y = (sum_n h*C + u*D) * silu(z).
// One thread per (b, d, n); 16 lanes (one state group) reduce via shfl_xor.
// ---------------------------------------------------------------------------
__global__ __launch_bounds__(256)
void scan_fused(const float* __restrict__ u, const float* __restrict__ delta_pre,
                const float* __restrict__ x_dbl, const float* __restrict__ xz,
                const float* __restrict__ A_log, const float* __restrict__ dt_b,
                const float* __restrict__ Dp, float* __restrict__ y)
{
    const int tid = blockIdx.x * blockDim.x + threadIdx.x;
    if (tid >= BB * D_INNER * D_STATE) return;
    const int n = tid & (D_STATE - 1);
    const int d = (tid >> 4) % D_INNER;
    const int b = tid / (D_STATE * D_INNER);

    const float An  = -__expf(A_log[d * D_STATE + n]);
    const float dtb = dt_b[d];
    const float Dd  = Dp[d];
    float h = 0.f;

    for (int t = 0; t < LL; ++t) {
        const size_t base = (size_t)(b * LL + t);
        const float dp = delta_pre[base * D_INNER + d];
        const float uu = u[base * D_INNER + d];
        const float bn = x_dbl[base * (DT_RANK + 2 * D_STATE) + DT_RANK + n];
        const float cn = x_dbl[base * (DT_RANK + 2 * D_STATE) + DT_RANK + D_STATE + n];

        const float xv  = dp + dtb;
        const float dlt = (xv > 20.f) ? xv : log1pf(__expf(xv));
        h = __expf(dlt * An) * h + dlt * uu * bn;
        float yn = h * cn;
#pragma unroll
        for (int off = 8; off > 0; off >>= 1)
            yn += __shfl_xor(yn, off, 16);
        if (n == 0) {
            const float z   = xz[base * (2 * D_INNER) + D_INNER + d];
            const float sil = z / (1.f + __expf(-z));
            y[base * D_INNER + d] = (yn + uu * Dd) * sil;
        }
    }
}

// ---------------------------------------------------------------------------
// Residual add + LayerNorm over D_MODEL=768 (one block per row).
// ---------------------------------------------------------------------------
__global__ __launch_bounds__(256)
void add_layernorm(const float* __restrict__ resid, const float* __restrict__ m,
                   const float* __restrict__ g, const float* __restrict__ beta,
                   float* __restrict__ dst)
{
    const int row = blockIdx.x;
    const int tid = threadIdx.x;
    float vals[3];
    float s = 0.f, s2 = 0.f;
#pragma unroll
    for (int i = 0; i < 3; ++i) {
        const int c = tid + i * 256;
        const float v = resid[(size_t)row * D_MODEL + c] +
                        m[(size_t)row * D_MODEL + c];
        vals[i] = v; s += v; s2 += v * v;
    }
    __shared__ float sh[256], sh2[256];
    sh[tid] = s; sh2[tid] = s2;
    __syncthreads();
    for (int off = 128; off > 0; off >>= 1) {
        if (tid < off) { sh[tid] += sh[tid + off]; sh2[tid] += sh2[tid + off]; }
        __syncthreads();
    }
    const float mu  = sh[0] * (1.f / D_MODEL);
    const float var = sh2[0] * (1.f / D_MODEL) - mu * mu;
    const float inv = rsqrtf(var + 1e-6f);
#pragma unroll
    for (int i = 0; i < 3; ++i) {
        const int c = tid + i * 256;
        dst[(size_t)row * D_MODEL + c] = (vals[i] - mu) * inv * g[c] + beta[c];
    }
}

// ---------------------------------------------------------------------------
extern "C" void kernel_launch(void* const* d_in, const int* in_sizes, int n_in,
                              void* d_out, int out_size, void* d_ws, size_t ws_size,
                              hipStream_t stream)
{
    (void)in_sizes; (void)n_in; (void)out_size; (void)ws_size;

    const float* x        = (const float*)d_in[0];
    const float* in_w     = (const float*)d_in[1];
    const float* conv_w   = (const float*)d_in[2];
    const float* conv_b   = (const float*)d_in[3];
    const float* xp_w     = (const float*)d_in[4];
    const float* dt_w     = (const float*)d_in[5];
    const float* dt_b     = (const float*)d_in[6];
    const float* A_log    = (const float*)d_in[7];
    const float* Dp       = (const float*)d_in[8];
    const float* out_w    = (const float*)d_in[9];
    const float* ln_g     = (const float*)d_in[10];
    const float* ln_b     = (const float*)d_in[11];

    float* ws    = (float*)d_ws;
    float* h     = ws;                                    // 2048*768
    float* xz    = h     + (size_t)M_ROWS * D_MODEL;      // 2048*3072
    float* u     = xz    + (size_t)M_ROWS * 2 * D_INNER;  // 2048*1536
    float* xdbl  = u     + (size_t)M_ROWS * D_INNER;      // 2048*80
    float* delta = xdbl  + (size_t)M_ROWS * (DT_RANK + 2 * D_STATE); // 2048*1536
    float* y     = delta + (size_t)M_ROWS * D_INNER;      // 2048*1536
    float* mout  = y     + (size_t)M_ROWS * D_INNER;      // 2048*768

    hipMemcpyAsync(h, x, (size_t)M_ROWS * D_MODEL * sizeof(float),
                   hipMemcpyDeviceToDevice, stream);

    const unsigned gy2 = (unsigned)(M_ROWS / 32);         // MT = 2 -> 64
    for (int i = 0; i < DEPTH; ++i) {
        const float* inw_i = in_w   + (size_t)i * 2 * D_INNER * D_MODEL;
        const float* cw_i  = conv_w + (size_t)i * D_INNER * D_CONV;
        const float* cb_i  = conv_b + (size_t)i * D_INNER;
        const float* xpw_i = xp_w   + (size_t)i * (DT_RANK + 2 * D_STATE) * D_INNER;
        const float* dtw_i = dt_w   + (size_t)i * D_INNER * DT_RANK;
        const float* dtb_i = dt_b   + (size_t)i * D_INNER;
        const float* Al_i  = A_log  + (size_t)i * D_INNER * D_STATE;
        const float* Dp_i  = Dp     + (size_t)i * D_INNER;
        const float* ow_i  = out_w  + (size_t)i * D_MODEL * D_INNER;
        const float* lg_i  = ln_g   + (size_t)i * D_MODEL;
        const float* lb_i  = ln_b   + (size_t)i * D_MODEL;

        // xz = h @ in_proj_w^T          [2048, 3072], K=768
        gemm_wmma_f32<4, 2><<<dim3(2 * D_INNER / 64, gy2), 32, 0, stream>>>(
            h, inw_i, xz, D_MODEL, D_MODEL, D_MODEL, 2 * D_INNER);

        // u = silu(depthwise_conv(xz[:, :1536]) + cb)
        conv_silu<<<(BB * LL * D_INNER + 255) / 256, 256, 0, stream>>>(
            xz, cw_i, cb_i, u);

        // x_dbl = u @ x_proj_w^T        [2048, 80], K=1536  (N=80 -> 5x16)
        gemm_wmma_f32<1, 2><<<dim3((DT_RANK + 2 * D_STATE) / 16, gy2), 32, 0, stream>>>(
            u, xpw_i, xdbl, D_INNER, D_INNER, D_INNER, DT_RANK + 2 * D_STATE);

        // delta_pre = x_dbl[:, :48] @ dt_proj_w^T   [2048, 1536], K=48
        gemm_wmma_f32<4, 2><<<dim3(D_INNER / 64, gy2), 32, 0, stream>>>(
            xdbl, dtw_i, delta, DT_RANK,
            DT_RANK + 2 * D_STATE, DT_RANK, D_INNER);

        // fused selective scan + gate
        scan_fused<<<(BB * D_INNER * D_STATE + 255) / 256, 256, 0, stream>>>(
            u, delta, xdbl, xz, Al_i, dtb_i, Dp_i, y);

        // mout = y @ out_proj_w^T       [2048, 768], K=1536
        gemm_wmma_f32<4, 2><<<dim3(D_MODEL / 64, gy2), 32, 0, stream>>>(
            y, ow_i, mout, D_INNER, D_INNER, D_INNER, D_MODEL);

        // h = layernorm(h + mout)
        float* dst = (i == DEPTH - 1) ? (float*)d_out : h;
        add_layernorm<<<M_ROWS, 256, 0, stream>>>(h, mout, lg_i, lb_i, dst);
    }
}